// SDT_35158602285468
// MI455X (gfx1250) — compile-verified
//
#include <hip/hip_runtime.h>
#include <math.h>

// SDT forward: B=16384, D=512, DEPTH=10, N=1023 inner nodes, L=1024 leaves, O=256
#define BB    16384
#define DD    512
#define DP1   513      // D+1 (bias feature prepended)
#define KP    516      // DP1 padded up to multiple of 4
#define NN    1023     // inner nodes
#define NP    1024     // padded node count
#define LL    1024     // leaves
#define OO    256
#define DEPTH 10
#define LAMDA 1e-3f
#define EPSV  1e-7f

typedef __attribute__((ext_vector_type(2))) float v2f;
typedef __attribute__((ext_vector_type(8))) float v8f;

#define WMMA_F32(a, b, c) \
    __builtin_amdgcn_wmma_f32_16x16x4_f32(false, (a), false, (b), (short)0, (c), false, false)

// ---------------------------------------------------------------------------
// Xaug[b][c]: c==0 -> 1.0 ; 1..512 -> data[b][c-1] ; 513..515 -> 0
__global__ void sdt_build_xaug(const float* __restrict__ data, float* __restrict__ X) {
    const int i = blockIdx.x * 256 + threadIdx.x;
    if (i >= BB * KP) return;
    const int b = i / KP, c = i % KP;
    float v;
    if (c == 0)          v = 1.0f;
    else if (c < DP1)    v = data[b * DD + (c - 1)];
    else                 v = 0.0f;
    X[i] = v;
}

// Wpad[r][c] = W[r][c] for r<1023 && c<513, else 0.  [1024, 516]
__global__ void sdt_build_wpad(const float* __restrict__ W, float* __restrict__ Wp) {
    const int i = blockIdx.x * 256 + threadIdx.x;
    if (i >= NP * KP) return;
    const int r = i / KP, c = i % KP;
    Wp[i] = (r < NN && c < DP1) ? W[r * DP1 + c] : 0.0f;
}

// ---------------------------------------------------------------------------
// GEMM1: P = sigmoid(beta * (Xaug @ Wpad^T)), P is [B, 1023] row-major.
// One wave -> 32x32 tile: 4 independent WMMA f32 16x16x4 accumulator chains.
// Tiles: 512 (M/32) x 32 (N/32) = 16384 waves.
__global__ void sdt_gemm1_sigmoid(const float* __restrict__ X,
                                  const float* __restrict__ Wp,
                                  const float* __restrict__ beta,
                                  float* __restrict__ P) {
    const int lane = threadIdx.x & 31;
    const int wib  = threadIdx.x >> 5;
    const int wid  = blockIdx.x * 8 + wib;
    const int NT   = 32;                 // N super-tiles
    const int mt   = wid / NT;
    const int nt   = wid % NT;
    const int m0   = mt * 32;
    const int n0   = nt * 32;
    const int half = lane >> 4;          // 0: K=2h..2h+1 with h=0 ; 1: K=2,3
    const int row  = lane & 15;

    const float* xr0 = X  + (size_t)(m0 + row)      * KP;
    const float* xr1 = X  + (size_t)(m0 + 16 + row) * KP;
    const float* wr0 = Wp + (size_t)(n0 + row)      * KP;
    const float* wr1 = Wp + (size_t)(n0 + 16 + row) * KP;

    v8f c00 = {}, c01 = {}, c10 = {}, c11 = {};
    for (int k = 0; k < KP; k += 4) {
        const int ka = k + 2 * half;
        const v2f a0 = *(const v2f*)(xr0 + ka);
        const v2f a1 = *(const v2f*)(xr1 + ka);
        const v2f b0 = *(const v2f*)(wr0 + ka);
        const v2f b1 = *(const v2f*)(wr1 + ka);
        c00 = WMMA_F32(a0, b0, c00);
        c01 = WMMA_F32(a0, b1, c01);
        c10 = WMMA_F32(a1, b0, c10);
        c11 = WMMA_F32(a1, b1, c11);
    }

    const int ncol0 = n0 + row;          // column for c00 / c10
    const int ncol1 = n0 + 16 + row;     // column for c01 / c11
    const float bt0 = (ncol0 < NN) ? beta[ncol0] : 0.0f;
    const float bt1 = (ncol1 < NN) ? beta[ncol1] : 0.0f;
    #pragma unroll
    for (int r = 0; r < 8; ++r) {
        const int mA = m0 + r + 8 * half;       // rows for c00/c01
        const int mB = mA + 16;                 // rows for c10/c11
        if (ncol0 < NN) {
            P[mA * NN + ncol0] = 1.0f / (1.0f + __expf(-bt0 * c00[r]));
            P[mB * NN + ncol0] = 1.0f / (1.0f + __expf(-bt0 * c10[r]));
        }
        if (ncol1 < NN) {
            P[mA * NN + ncol1] = 1.0f / (1.0f + __expf(-bt1 * c01[r]));
            P[mB * NN + ncol1] = 1.0f / (1.0f + __expf(-bt1 * c11[r]));
        }
    }
}

// ---------------------------------------------------------------------------
// mu leaves: one block per batch row; stage P row in LDS; walk heap paths.
// Fused argmax over leaves -> ids[b].
__global__ void sdt_mu_leaves(const float* __restrict__ P,
                              float* __restrict__ MU,
                              int* __restrict__ IDS) {
    __shared__ float sp[NN];
    __shared__ float smax[256];
    __shared__ int   sidx[256];
    const int b = blockIdx.x;
    const int t = threadIdx.x;

    for (int i = t; i < NN; i += 256) sp[i] = P[b * NN + i];
    __syncthreads();

    float bestv = -1.0f;
    int   besti = 0;
    #pragma unroll
    for (int u = 0; u < 4; ++u) {
        const int leaf = u * 256 + t;
        float m = 1.0f;
        int node = 0;                       // heap index: children 2i+1, 2i+2
        #pragma unroll
        for (int layer = 0; layer < DEPTH; ++layer) {
            const int bit = (leaf >> (DEPTH - 1 - layer)) & 1;
            const float pv = sp[node];
            m *= bit ? (1.0f - pv) : pv;    // left (bit 0) = p, right = 1-p
            node = 2 * node + 1 + bit;
        }
        MU[b * LL + leaf] = m;
        if (m > bestv || (m == bestv && leaf < besti)) { bestv = m; besti = leaf; }
    }
    smax[t] = bestv; sidx[t] = besti;
    __syncthreads();
    for (int s = 128; s > 0; s >>= 1) {
        if (t < s) {
            const float ov = smax[t + s]; const int oi = sidx[t + s];
            if (ov > smax[t] || (ov == smax[t] && oi < sidx[t])) { smax[t] = ov; sidx[t] = oi; }
        }
        __syncthreads();
    }
    if (t == 0) IDS[b] = sidx[0];
}

// ---------------------------------------------------------------------------
__global__ void sdt_zero_S(float* __restrict__ S) {
    const int i = blockIdx.x * 256 + threadIdx.x;
    if (i < 2048) S[i] = 0.0f;
}

// Column sums of MU into heap leaf slots S[1023 + leaf].
// grid = (64 leaf tiles of 16, 64 batch chunks of 256), block = 256.
__global__ void sdt_colsum(const float* __restrict__ MU, float* __restrict__ S) {
    const int t    = threadIdx.x;
    const int leaf = blockIdx.x * 16 + (t & 15);
    const int rl   = t >> 4;
    const int b0   = blockIdx.y * 256;
    float s = 0.0f;
    #pragma unroll
    for (int i = 0; i < 16; ++i) s += MU[(size_t)(b0 + rl + i * 16) * LL + leaf];
    __shared__ float red[256];
    red[t] = s; __syncthreads();
    for (int st = 128; st >= 16; st >>= 1) {
        if (t < st) red[t] += red[t + st];
        __syncthreads();
    }
    if (t < 16) atomicAdd(&S[NN + blockIdx.x * 16 + t], red[t]);
}

// ---------------------------------------------------------------------------
// Penalty: single block. Build internal heap sums, then accumulate the
// layer-weighted log penalty. S has 2047 heap entries (root=0).
__global__ void sdt_penalty(float* __restrict__ S, float* __restrict__ out_pen) {
    const int t = threadIdx.x;
    for (int L = DEPTH - 1; L >= 0; --L) {
        const int n = 1 << L, base = n - 1;
        if (t < n) {
            const int i = base + t;
            S[i] = S[2 * i + 1] + S[2 * i + 2];
        }
        __syncthreads();
    }
    float acc = 0.0f;
    for (int L = 0; L < DEPTH; ++L) {
        const int slots = 2 << L;                    // 2n
        const float coeff = LAMDA * exp2f(-(float)L) * 0.5f;
        for (int s = t; s < slots; s += blockDim.x) {
            const float denom = S[(1 << L) - 1 + (s >> 1)] + EPSV;
            float alpha = S[(2 << L) - 1 + s] / denom;
            alpha = fminf(fmaxf(alpha, EPSV), 1.0f - EPSV);
            acc -= coeff * (__logf(alpha) + __logf(1.0f - alpha));
        }
    }
    __shared__ float red[1024];
    red[t] = acc; __syncthreads();
    for (int st = 512; st > 0; st >>= 1) {
        if (t < st) red[t] += red[t + st];
        __syncthreads();
    }
    if (t == 0) *out_pen = red[0];
}

// ---------------------------------------------------------------------------
// Row softmax of param -> distT (transposed [O, L]) and log-dist [L, O].
// One block (256 thr) per row.
__global__ void sdt_softmax_rows(const float* __restrict__ param,
                                 float* __restrict__ distT,
                                 float* __restrict__ logd) {
    const int r = blockIdx.x, t = threadIdx.x;
    const float v = param[r * OO + t];
    __shared__ float red[256];
    red[t] = v; __syncthreads();
    for (int s = 128; s > 0; s >>= 1) { if (t < s) red[t] = fmaxf(red[t], red[t + s]); __syncthreads(); }
    const float mx = red[0];
    __syncthreads();
    const float e = __expf(v - mx);
    red[t] = e; __syncthreads();
    for (int s = 128; s > 0; s >>= 1) { if (t < s) red[t] += red[t + s]; __syncthreads(); }
    const float sum = red[0];
    distT[(size_t)t * LL + r] = e / sum;            // B-operand layout for GEMM2
    logd[r * OO + t] = v - mx - __logf(sum);
}

// ---------------------------------------------------------------------------
// GEMM2: out1 = log(MU @ dist).  16384x1024x256 with distT [O, L].
// One wave -> 32x32 tile, 4 accumulators. Tiles: 512 (M/32) x 8 (N/32).
__global__ void sdt_gemm2_log(const float* __restrict__ MU,
                              const float* __restrict__ distT,
                              float* __restrict__ out1) {
    const int lane = threadIdx.x & 31;
    const int wib  = threadIdx.x >> 5;
    const int wid  = blockIdx.x * 8 + wib;
    const int NT   = 8;
    const int mt   = wid / NT;
    const int nt   = wid % NT;
    const int m0   = mt * 32;
    const int n0   = nt * 32;
    const int half = lane >> 4;
    const int row  = lane & 15;

    const float* ar0 = MU    + (size_t)(m0 + row)      * LL;
    const float* ar1 = MU    + (size_t)(m0 + 16 + row) * LL;
    const float* br0 = distT + (size_t)(n0 + row)      * LL;   // column n0+row of dist
    const float* br1 = distT + (size_t)(n0 + 16 + row) * LL;

    v8f c00 = {}, c01 = {}, c10 = {}, c11 = {};
    for (int k = 0; k < LL; k += 4) {
        const int ka = k + 2 * half;
        const v2f a0 = *(const v2f*)(ar0 + ka);
        const v2f a1 = *(const v2f*)(ar1 + ka);
        const v2f b0 = *(const v2f*)(br0 + ka);
        const v2f b1 = *(const v2f*)(br1 + ka);
        c00 = WMMA_F32(a0, b0, c00);
        c01 = WMMA_F32(a0, b1, c01);
        c10 = WMMA_F32(a1, b0, c10);
        c11 = WMMA_F32(a1, b1, c11);
    }

    const int n0c = n0 + row;
    const int n1c = n0 + 16 + row;
    #pragma unroll
    for (int r = 0; r < 8; ++r) {
        const int mA = m0 + r + 8 * half;
        const int mB = mA + 16;
        out1[(size_t)mA * OO + n0c] = __logf(c00[r]);
        out1[(size_t)mA * OO + n1c] = __logf(c01[r]);
        out1[(size_t)mB * OO + n0c] = __logf(c10[r]);
        out1[(size_t)mB * OO + n1c] = __logf(c11[r]);
    }
}

// ---------------------------------------------------------------------------
// prediction log-prob: out0[b] = logd[ids[b]]; one block (256 thr) per b.
__global__ void sdt_gather_pred(const float* __restrict__ logd,
                                const int* __restrict__ IDS,
                                float* __restrict__ out0) {
    const int b = blockIdx.x, t = threadIdx.x;
    out0[(size_t)b * OO + t] = logd[(size_t)IDS[b] * OO + t];
}

__global__ void sdt_copy_w(const float* __restrict__ W, float* __restrict__ outW, int n) {
    const int i = blockIdx.x * 256 + threadIdx.x;
    if (i < n) outW[i] = W[i];
}

// ---------------------------------------------------------------------------
extern "C" void kernel_launch(void* const* d_in, const int* in_sizes, int n_in,
                              void* d_out, int out_size, void* d_ws, size_t ws_size,
                              hipStream_t stream) {
    const float* data  = (const float*)d_in[0];  // [B, 512]
    const float* W     = (const float*)d_in[1];  // [1023, 513]
    const float* beta  = (const float*)d_in[2];  // [1023]
    const float* param = (const float*)d_in[3];  // [1024, 256]

    float* out  = (float*)d_out;
    float* out0 = out;                         // log(prediction) [B, 256]
    float* out1 = out + (size_t)BB * OO;       // log(output)     [B, 256]
    float* outP = out + 2 * (size_t)BB * OO;   // penalty scalar
    float* outW = outP + 1;                    // W copy [1023, 513]

    // workspace layout (floats)
    float* ws    = (float*)d_ws;
    float* XAUG  = ws;                                // 16384*516
    float* WPAD  = XAUG  + (size_t)BB * KP;           // 1024*516
    float* P     = WPAD  + (size_t)NP * KP;           // 16384*1023
    float* MU    = P     + (size_t)BB * NN;           // 16384*1024
    float* DISTT = MU    + (size_t)BB * LL;           // 256*1024 (dist^T)
    float* LOGD  = DISTT + (size_t)OO * LL;           // 1024*256
    float* S     = LOGD  + (size_t)LL * OO;           // 2048 heap sums
    int*   IDS   = (int*)(S + 2048);                  // 16384

    // 0) operand prep + zero heap sums
    sdt_build_xaug<<<(BB * KP + 255) / 256, 256, 0, stream>>>(data, XAUG);
    sdt_build_wpad<<<(NP * KP + 255) / 256, 256, 0, stream>>>(W, WPAD);
    sdt_zero_S<<<8, 256, 0, stream>>>(S);

    // 1) P = sigmoid(beta * (x @ W^T)) : 512x32 super-tiles, 8 waves/block
    sdt_gemm1_sigmoid<<<(512 * 32) / 8, 256, 0, stream>>>(XAUG, WPAD, beta, P);

    // 2) mu leaves + argmax ids
    sdt_mu_leaves<<<BB, 256, 0, stream>>>(P, MU, IDS);

    // 3) leaf column sums
    sdt_colsum<<<dim3(64, 64), 256, 0, stream>>>(MU, S);

    // 4) softmax(param) -> distT, logdist
    sdt_softmax_rows<<<LL, 256, 0, stream>>>(param, DISTT, LOGD);

    // 5) penalty scalar
    sdt_penalty<<<1, 1024, 0, stream>>>(S, outP);

    // 6) log(mu @ dist) : 512x8 super-tiles, 8 waves/block
    sdt_gemm2_log<<<(512 * 8) / 8, 256, 0, stream>>>(MU, DISTT, out1);

    // 7) log(prediction) via hard argmax gather
    sdt_gather_pred<<<BB, 256, 0, stream>>>(LOGD, IDS, out0);

    // 8) pass-through W
    sdt_copy_w<<<(NN * DP1 + 255) / 256, 256, 0, stream>>>(W, outW, NN * DP1);
}